// SLOG_B_56014963474691
// MI455X (gfx1250) — compile-verified
//
#include <hip/hip_runtime.h>
#include <math.h>

typedef __attribute__((ext_vector_type(2))) float v2f;
typedef __attribute__((ext_vector_type(8))) float v8f;

#define NROW 8192
#define CDIM 64
#define KB   32      // K chunk per LDS stage
#define MT   128     // M tile per workgroup
#define SPLITK 2
#define KSEG (NROW / SPLITK)     // 4096 K per block
// LDS layouts (dword strides):
//  TRANS=1 A tile: [k/2][m][2], stride SA_T per k-pair row; 288 % 64 == 32 -> the
//  two lane-halves (k-pair vs k-pair+1) hit disjoint 32-bank sets (conflict-free b64).
#define SA_T 288     // 16 k-pair rows x 288 = 4608
//  TRANS=0 A tile: [m][k], {k,k+1} contiguous; 36*l banks all distinct mod 64.
#define SA_N 36      // 128 rows x 36 = 4608
//  B tile: [k/2][n][2], stride 96 (96 % 64 == 32 -> conflict-free b64).
#define SBP  96      // 16 k-pair rows x 96 = 1536
#define USZ  4608
#define BSZ  1536

// ---------------------------------------------------------------------------
// Spectral GEMM partials (split-K):
//   TRANS=1: P[kz] = (U^T @ In) over K segment kz;  TRANS=0: P[kz] = (U @ In) seg kz
// U: [8192,8192] row-major, In: [8192,64], Pout: SPLITK x [8192,64].
// 8 waves: wave w owns M strips {2(w&3), 2(w&3)+1} x N half (w>>2):
// 2x2 = 4 accumulators, 4 b64 LDS loads : 4 independent WMMAs per K micro-step.
// ---------------------------------------------------------------------------
template <int TRANS>
__launch_bounds__(256, 2)
__global__ void spectral_gemm(const float* __restrict__ U,
                              const float* __restrict__ In,
                              float* __restrict__ Pout)
{
    __shared__ __align__(16) float uS[2][USZ];
    __shared__ __align__(16) float bS[2][BSZ];

    const int tid  = threadIdx.x;
    const int wv   = tid >> 5;        // 8 waves
    const int lane = tid & 31;
    const int l15  = lane & 15;
    const int hi   = lane >> 4;       // lane half: K pair (A/B), M half (C)
    const int m0   = blockIdx.x * MT;
    const int kb0  = blockIdx.y * KSEG;
    float* __restrict__ P = Pout + (size_t)blockIdx.y * NROW * CDIM;

    const int mloc = (wv & 3) * 32;   // wave's pair of 16-row strips
    const int noff = (wv >> 2) * 32;  // wave's 32-column half

    v8f c[4] = {v8f{}, v8f{}, v8f{}, v8f{}};   // [strip][ntile]

    float4 ru0, ru1, ru2, ru3, rb0, rb1;       // global->LDS staging registers

    auto gload = [&](int k0) {
#pragma unroll
        for (int r = 0; r < 4; ++r) {
            const int i = tid + 256 * r;
            float4 v;
            if (TRANS) {
                const int kk = i >> 5, c4 = i & 31;          // 32 rows x 32 float4
                v = *(const float4*)(U + (size_t)(k0 + kk) * NROW + m0 + c4 * 4);
            } else {
                const int mm = i >> 3, c4 = i & 7;           // 128 rows x 8 float4
                v = *(const float4*)(U + (size_t)(m0 + mm) * NROW + k0 + c4 * 4);
            }
            if      (r == 0) ru0 = v;
            else if (r == 1) ru1 = v;
            else if (r == 2) ru2 = v;
            else             ru3 = v;
        }
#pragma unroll
        for (int r = 0; r < 2; ++r) {
            const int i = tid + 256 * r;
            const int kk = i >> 4, c4 = i & 15;              // 32 rows x 16 float4
            float4 v = *(const float4*)(In + (size_t)(k0 + kk) * CDIM + c4 * 4);
            if (r == 0) rb0 = v; else rb1 = v;
        }
    };

    auto sstore = [&](int buf) {
#pragma unroll
        for (int r = 0; r < 4; ++r) {
            const int i = tid + 256 * r;
            const float4 v = (r == 0) ? ru0 : (r == 1) ? ru1 : (r == 2) ? ru2 : ru3;
            if (TRANS) {
                const int kk = i >> 5, c4 = i & 31;
                const int base = (kk >> 1) * SA_T + (kk & 1);
#pragma unroll
                for (int j = 0; j < 4; ++j)
                    uS[buf][base + (c4 * 4 + j) * 2] = ((const float*)&v)[j];
            } else {
                const int mm = i >> 3, c4 = i & 7;
                *(float4*)(&uS[buf][mm * SA_N + c4 * 4]) = v;
            }
        }
#pragma unroll
        for (int r = 0; r < 2; ++r) {
            const int i = tid + 256 * r;
            const int kk = i >> 4, c4 = i & 15;
            const float4 v = (r == 0) ? rb0 : rb1;
            const int base = (kk >> 1) * SBP + (kk & 1);
#pragma unroll
            for (int j = 0; j < 4; ++j)
                bS[buf][base + (c4 * 4 + j) * 2] = ((const float*)&v)[j];
        }
    };

    // fragment fetch for one K micro-step (K=4 starting at kk within the chunk);
    // kin = kk + 2*hi is even, so every v2f is one aligned 8-byte LDS load.
    auto lfrag = [&](int cur, int kk, v2f a[2], v2f b[2]) {
        const int kin  = kk + hi * 2;
        const int krow = kin >> 1;
#pragma unroll
        for (int s = 0; s < 2; ++s) {
            const int m = mloc + s * 16 + l15;
            if (TRANS)
                a[s] = *(const v2f*)(&uS[cur][krow * SA_T + m * 2]);
            else
                a[s] = *(const v2f*)(&uS[cur][m * SA_N + kin]);
        }
#pragma unroll
        for (int t = 0; t < 2; ++t) {
            const int n = noff + t * 16 + l15;
            b[t] = *(const v2f*)(&bS[cur][krow * SBP + n * 2]);
        }
    };

    gload(kb0);
    sstore(0);

    const int NCH = KSEG / KB;  // 128 chunks
    for (int t = 0; t < NCH; ++t) {
        __syncthreads();
        if (t + 1 < NCH) gload(kb0 + (t + 1) * KB);
        const int cur = t & 1;

        v2f aF[2][2], bF[2][2];
        lfrag(cur, 0, aF[0], bF[0]);
#pragma unroll
        for (int s = 0; s < KB / 4; ++s) {        // 8 micro-steps
            const int p = s & 1;
            if (s + 1 < KB / 4) lfrag(cur, (s + 1) * 4, aF[p ^ 1], bF[p ^ 1]);
            c[0] = __builtin_amdgcn_wmma_f32_16x16x4_f32(
                false, aF[p][0], false, bF[p][0], (short)0, c[0], false, false);
            c[1] = __builtin_amdgcn_wmma_f32_16x16x4_f32(
                false, aF[p][0], false, bF[p][1], (short)0, c[1], false, false);
            c[2] = __builtin_amdgcn_wmma_f32_16x16x4_f32(
                false, aF[p][1], false, bF[p][0], (short)0, c[2], false, false);
            c[3] = __builtin_amdgcn_wmma_f32_16x16x4_f32(
                false, aF[p][1], false, bF[p][1], (short)0, c[3], false, false);
        }
        __syncthreads();
        if (t + 1 < NCH) sstore((t + 1) & 1);
    }

    // Epilogue: C layout -> VGPR r holds M = strip_base + hi*8 + r, N = l15 in tile
#pragma unroll
    for (int s = 0; s < 2; ++s) {
#pragma unroll
        for (int t = 0; t < 2; ++t) {
            const v8f& acc = c[s * 2 + t];
#pragma unroll
            for (int r = 0; r < 8; ++r) {
                const int mg = m0 + mloc + s * 16 + hi * 8 + r;
                P[(size_t)mg * CDIM + noff + t * 16 + l15] = acc[r];
            }
        }
    }
}

// ---------------------------------------------------------------------------
// Deterministic split-K reduction + optional spectral row-scale.
// Out[m][n] = (P0[m][n] + P1[m][n]) * (rowscale ? rowscale[m] : 1)
// ---------------------------------------------------------------------------
__launch_bounds__(256)
__global__ void reduce_scale(const float* __restrict__ P0,
                             const float* __restrict__ P1,
                             const float* __restrict__ rowscale,
                             float* __restrict__ Out)
{
    const int idx = blockIdx.x * 256 + threadIdx.x;      // float4 index
    const int m = idx >> 4;                              // 16 float4 per row
    const float s = rowscale ? rowscale[m] : 1.0f;
    const float4 a = ((const float4*)P0)[idx];
    const float4 b = ((const float4*)P1)[idx];
    float4 o;
    o.x = (a.x + b.x) * s;
    o.y = (a.y + b.y) * s;
    o.z = (a.z + b.z) * s;
    o.w = (a.w + b.w) * s;
    ((float4*)Out)[idx] = o;
}

// ---------------------------------------------------------------------------
// Eigenvalue scale vectors: vs1 = La^ve, vs2 = ((2(La-eps)-1)^2 + 1)^ve2
// ---------------------------------------------------------------------------
__global__ void scales_kernel(const float* __restrict__ La,
                              const float* __restrict__ ve,
                              const float* __restrict__ ve2,
                              float* __restrict__ vs1, float* __restrict__ vs2)
{
    const int i = blockIdx.x * 256 + threadIdx.x;
    if (i < NROW) {
        const float la = La[i];
        vs1[i] = powf(la, ve[0]);
        const float t = 2.0f * (la - 1e-8f) - 1.0f;
        vs2[i] = powf(t * t + 1.0f, ve2[0]);
    }
}

__global__ void zero_kernel(float* __restrict__ p, int n)
{
    const int i = blockIdx.x * 128 + threadIdx.x;
    if (i < n) p[i] = 0.0f;
}

// ---------------------------------------------------------------------------
// hidden = F @ W^T + b  (64x64), plus BN sum / sumsq accumulation per column.
// ---------------------------------------------------------------------------
__launch_bounds__(256)
__global__ void linbn_kernel(const float* __restrict__ F,
                             const float* __restrict__ W,
                             const float* __restrict__ Wb,
                             float* __restrict__ hidden,
                             float* __restrict__ bnsum,
                             float* __restrict__ bnsq)
{
    __shared__ float Ws[64 * 64];
    __shared__ float ssum[64], ssq[64];
    const int tid = threadIdx.x;
    for (int i = tid; i < 64 * 64; i += 256) Ws[i] = W[i];
    if (tid < 64) { ssum[tid] = 0.0f; ssq[tid] = 0.0f; }
    __syncthreads();

    const int m = blockIdx.x * 256 + tid;
    float x[64];
#pragma unroll
    for (int i = 0; i < 64; ++i) x[i] = F[(size_t)m * 64 + i];

    for (int o = 0; o < 64; ++o) {
        float acc = Wb[o];
#pragma unroll
        for (int i = 0; i < 64; ++i) acc = fmaf(x[i], Ws[o * 64 + i], acc);
        hidden[(size_t)m * 64 + o] = acc;
        atomicAdd(&ssum[o], acc);
        atomicAdd(&ssq[o], acc * acc);
    }
    __syncthreads();
    if (tid < 64) {
        atomicAdd(&bnsum[tid], ssum[tid]);
        atomicAdd(&bnsq[tid],  ssq[tid]);
    }
}

// ---------------------------------------------------------------------------
// BN-normalize + ReLU + (64 -> 32) MLP + log_softmax.
// ---------------------------------------------------------------------------
__launch_bounds__(256)
__global__ void tail_kernel(const float* __restrict__ hidden,
                            const float* __restrict__ bnsum,
                            const float* __restrict__ bnsq,
                            const float* __restrict__ gamma,
                            const float* __restrict__ beta,
                            const float* __restrict__ mw,
                            const float* __restrict__ mb,
                            float* __restrict__ out)
{
    __shared__ float Ms[32 * 64];
    __shared__ float meanS[64], rstdS[64], gS[64], btS[64];
    const int tid = threadIdx.x;
    for (int i = tid; i < 32 * 64; i += 256) Ms[i] = mw[i];
    if (tid < 64) {
        const float mu  = bnsum[tid] * (1.0f / NROW);
        const float var = bnsq[tid] * (1.0f / NROW) - mu * mu;
        meanS[tid] = mu;
        rstdS[tid] = rsqrtf(var + 1e-5f);
        gS[tid]  = gamma[tid];
        btS[tid] = beta[tid];
    }
    __syncthreads();

    const int m = blockIdx.x * 256 + tid;
    float x[64];
#pragma unroll
    for (int i = 0; i < 64; ++i) {
        float h = hidden[(size_t)m * 64 + i];
        h = (h - meanS[i]) * rstdS[i] * gS[i] + btS[i];
        x[i] = fmaxf(h, 0.0f);
    }
    float y[32];
    for (int o = 0; o < 32; ++o) {
        float acc = mb[o];
#pragma unroll
        for (int i = 0; i < 64; ++i) acc = fmaf(x[i], Ms[o * 64 + i], acc);
        y[o] = acc;
    }
    float mx = -INFINITY;
#pragma unroll
    for (int o = 0; o < 32; ++o) mx = fmaxf(mx, y[o]);
    float se = 0.0f;
#pragma unroll
    for (int o = 0; o < 32; ++o) se += expf(y[o] - mx);
    const float lse = mx + logf(se);
#pragma unroll
    for (int o = 0; o < 32; ++o) out[(size_t)m * 32 + o] = y[o] - lse;
}

// ---------------------------------------------------------------------------
extern "C" void kernel_launch(void* const* d_in, const int* in_sizes, int n_in,
                              void* d_out, int out_size, void* d_ws, size_t ws_size,
                              hipStream_t stream)
{
    const float* X     = (const float*)d_in[0];
    const float* La    = (const float*)d_in[1];
    const float* U     = (const float*)d_in[2];
    const float* ve    = (const float*)d_in[3];
    const float* ve2   = (const float*)d_in[4];
    const float* Ww    = (const float*)d_in[5];
    const float* Wb    = (const float*)d_in[6];
    const float* gamma = (const float*)d_in[7];
    const float* beta  = (const float*)d_in[8];
    const float* mw    = (const float*)d_in[9];
    const float* mb    = (const float*)d_in[10];

    float* outLS  = (float*)d_out;                      // [8192,32] log_softmax
    float* hidden = (float*)d_out + (size_t)NROW * 32;  // [8192,64] hidden_emd

    const size_t ACT = (size_t)NROW * CDIM;             // 524288
    float* ws    = (float*)d_ws;
    float* vs1   = ws;                                  // 8192
    float* vs2   = ws + NROW;                           // 8192
    float* T0    = ws + 2 * NROW;                       // ACT
    float* T1    = T0 + ACT;                            // ACT
    float* P0    = T1 + ACT;                            // ACT (split-K partial 0)
    float* P1    = P0 + ACT;                            // ACT (split-K partial 1)
    float* bnsum = P1 + ACT;                            // 64
    float* bnsq  = bnsum + 64;                          // 64

    const dim3 ggrid(NROW / MT, SPLITK);                // 64 x 2 = 128 blocks
    const int rblocks = (NROW * CDIM / 4) / 256;        // 512 blocks (float4)

    scales_kernel<<<NROW / 256, 256, 0, stream>>>(La, ve, ve2, vs1, vs2);

    // out = U @ (vs2 ⊙ (U^T @ (U @ (vs1 ⊙ (U^T @ X)))))
    spectral_gemm<1><<<ggrid, 256, 0, stream>>>(U, X, P0);
    reduce_scale<<<rblocks, 256, 0, stream>>>(P0, P1, vs1, T0);

    spectral_gemm<0><<<ggrid, 256, 0, stream>>>(U, T0, P0);
    reduce_scale<<<rblocks, 256, 0, stream>>>(P0, P1, nullptr, T1);

    spectral_gemm<1><<<ggrid, 256, 0, stream>>>(U, T1, P0);
    reduce_scale<<<rblocks, 256, 0, stream>>>(P0, P1, vs2, T0);

    spectral_gemm<0><<<ggrid, 256, 0, stream>>>(U, T0, P0);
    reduce_scale<<<rblocks, 256, 0, stream>>>(P0, P1, nullptr, T1);

    zero_kernel<<<1, 128, 0, stream>>>(bnsum, 128);     // zeros bnsum + bnsq
    linbn_kernel<<<NROW / 256, 256, 0, stream>>>(T1, Ww, Wb, hidden, bnsum, bnsq);
    tail_kernel<<<NROW / 256, 256, 0, stream>>>(hidden, bnsum, bnsq,
                                                gamma, beta, mw, mb, outLS);
}